// GNN_78288663872123
// MI455X (gfx1250) — compile-verified
//
#include <hip/hip_runtime.h>

#define N_NODES 50000
#define E_EDGES 800000
#define F_IN    64
#define F_HID   128
#define F_OUT   8

typedef __attribute__((ext_vector_type(2))) float v2f;
typedef __attribute__((ext_vector_type(8))) float v8f;

// ---------------- degree / normalization ----------------

__global__ void GCN_init_deg(float* __restrict__ deg) {
    int i = blockIdx.x * blockDim.x + threadIdx.x;
    if (i < N_NODES) deg[i] = 1.0f;               // self-loop contributes 1
}

__global__ void GCN_count_deg(const int* __restrict__ dst, float* __restrict__ deg) {
    int e = blockIdx.x * blockDim.x + threadIdx.x;
    if (e < E_EDGES) atomicAdd(&deg[dst[e]], 1.0f);
}

__global__ void GCN_rsqrt(float* __restrict__ dinv) {
    int i = blockIdx.x * blockDim.x + threadIdx.x;
    if (i < N_NODES) dinv[i] = rsqrtf(dinv[i]);   // deg >= 1 always (self-loop)
}

// ---------------- GEMM1: h = x @ W1  [50000,64]x[64,128] via f32 WMMA ----------------
// One wave per 16x16 output tile. Block = 8 waves = one row-tile, 8 col-tiles.

__global__ void GCN_gemm1(const float* __restrict__ x, const float* __restrict__ W1,
                          float* __restrict__ h) {
    const int wave = threadIdx.x >> 5;      // 0..7 -> column tile
    const int lane = threadIdx.x & 31;
    const int half = lane >> 4;             // 0: K=0,1 ; 1: K=2,3
    const int l    = lane & 15;
    const int m0   = blockIdx.x * 16;       // 3125 blocks cover 50000 rows exactly
    const int n0   = wave * 16;

    v8f c = {};
    const int arow = m0 + l;
#pragma unroll
    for (int k0 = 0; k0 < F_IN; k0 += 4) {
        const int kb = k0 + half * 2;
        v2f a, b;
        // A 16x4: lane l holds row m0+l, K = kb, kb+1
        a.x = x[arow * F_IN + kb];
        a.y = x[arow * F_IN + kb + 1];
        // B 4x16: lane l holds col n0+l, K rows kb, kb+1
        b.x = W1[kb * F_HID + n0 + l];
        b.y = W1[(kb + 1) * F_HID + n0 + l];
        c = __builtin_amdgcn_wmma_f32_16x16x4_f32(false, a, false, b,
                                                  (short)0, c, false, false);
    }
#pragma unroll
    for (int r = 0; r < 8; ++r)
        h[(m0 + r + half * 8) * F_HID + n0 + l] = c[r];
}

// ---------------- layer-1 aggregation ----------------
// init: agg[i] = h[i] * dinv[i]^2   (self-loop term)

__global__ void GCN_self1(const float* __restrict__ h, const float* __restrict__ dinv,
                          float* __restrict__ agg) {
    int idx = blockIdx.x * blockDim.x + threadIdx.x;       // over N * (128/4)
    if (idx >= N_NODES * (F_HID / 4)) return;
    int i  = idx >> 5;                                     // 32 float4 per row
    int c4 = idx & 31;
    float w = dinv[i]; w = w * w;
    float4 hv = reinterpret_cast<const float4*>(h + (size_t)i * F_HID)[c4];
    float4 o; o.x = hv.x * w; o.y = hv.y * w; o.z = hv.z * w; o.w = hv.w * w;
    reinterpret_cast<float4*>(agg + (size_t)i * F_HID)[c4] = o;
}

// one wave per edge: float4 gather of h[src], 4 f32 atomics per lane into agg[dst]
__global__ void GCN_edge1(const int* __restrict__ src, const int* __restrict__ dst,
                          const float* __restrict__ dinv, const float* __restrict__ h,
                          float* __restrict__ agg) {
    int gid  = blockIdx.x * blockDim.x + threadIdx.x;
    int e    = gid >> 5;
    int lane = gid & 31;
    if (e >= E_EDGES) return;
    int s = src[e], d = dst[e];
    float norm = dinv[s] * dinv[d];
    float4 hv = reinterpret_cast<const float4*>(h + (size_t)s * F_HID)[lane];
    float* ap = agg + (size_t)d * F_HID + lane * 4;
    atomicAdd(ap + 0, hv.x * norm);
    atomicAdd(ap + 1, hv.y * norm);
    atomicAdd(ap + 2, hv.z * norm);
    atomicAdd(ap + 3, hv.w * norm);
}

// ---------------- GEMM2: h2 = relu(agg + b1) @ W2  [50000,128]x[128,8] ----------------
// bias + ReLU fused into A-operand load; B zero-padded from 8 to 16 columns.

__global__ void GCN_gemm2(const float* __restrict__ agg, const float* __restrict__ b1,
                          const float* __restrict__ W2, float* __restrict__ h2) {
    const int tile = blockIdx.x * 8 + (threadIdx.x >> 5);   // row tile (wave-uniform)
    if (tile >= N_NODES / 16) return;                       // whole-wave exit: EXEC stays all-1
    const int lane = threadIdx.x & 31;
    const int half = lane >> 4;
    const int l    = lane & 15;
    const int m0   = tile * 16;

    v8f c = {};
    const int arow = m0 + l;
    const bool valid_col = (l < F_OUT);
#pragma unroll
    for (int k0 = 0; k0 < F_HID; k0 += 4) {
        const int kb = k0 + half * 2;
        v2f a, b;
        a.x = fmaxf(agg[arow * F_HID + kb]     + b1[kb],     0.0f);
        a.y = fmaxf(agg[arow * F_HID + kb + 1] + b1[kb + 1], 0.0f);
        b.x = valid_col ? W2[kb * F_OUT + l]       : 0.0f;
        b.y = valid_col ? W2[(kb + 1) * F_OUT + l] : 0.0f;
        c = __builtin_amdgcn_wmma_f32_16x16x4_f32(false, a, false, b,
                                                  (short)0, c, false, false);
    }
    if (valid_col) {
#pragma unroll
        for (int r = 0; r < 8; ++r)
            h2[(m0 + r + half * 8) * F_OUT + l] = c[r];
    }
}

// ---------------- layer-2 aggregation (into d_out) ----------------
// init: out[i] = h2[i] * dinv[i]^2 + b2

__global__ void GCN_self2(const float* __restrict__ h2, const float* __restrict__ dinv,
                          const float* __restrict__ b2, float* __restrict__ out) {
    int idx = blockIdx.x * blockDim.x + threadIdx.x;
    if (idx >= N_NODES * F_OUT) return;
    int i = idx >> 3, ch = idx & 7;
    float w = dinv[i];
    out[idx] = h2[idx] * w * w + b2[ch];
}

// one thread per edge: 8 channels, two float4 gathers, 8 atomics
__global__ void GCN_edge2(const int* __restrict__ src, const int* __restrict__ dst,
                          const float* __restrict__ dinv, const float* __restrict__ h2,
                          float* __restrict__ out) {
    int e = blockIdx.x * blockDim.x + threadIdx.x;
    if (e >= E_EDGES) return;
    int s = src[e], d = dst[e];
    float norm = dinv[s] * dinv[d];
    float4 a = reinterpret_cast<const float4*>(h2 + (size_t)s * F_OUT)[0];
    float4 b = reinterpret_cast<const float4*>(h2 + (size_t)s * F_OUT)[1];
    float* op = out + (size_t)d * F_OUT;
    atomicAdd(op + 0, a.x * norm);
    atomicAdd(op + 1, a.y * norm);
    atomicAdd(op + 2, a.z * norm);
    atomicAdd(op + 3, a.w * norm);
    atomicAdd(op + 4, b.x * norm);
    atomicAdd(op + 5, b.y * norm);
    atomicAdd(op + 6, b.z * norm);
    atomicAdd(op + 7, b.w * norm);
}

// ---------------- launch ----------------

extern "C" void kernel_launch(void* const* d_in, const int* in_sizes, int n_in,
                              void* d_out, int out_size, void* d_ws, size_t ws_size,
                              hipStream_t stream) {
    const float* x  = (const float*)d_in[0];
    const int*   ei = (const int*)d_in[1];
    const float* W1 = (const float*)d_in[2];
    const float* b1 = (const float*)d_in[3];
    const float* W2 = (const float*)d_in[4];
    const float* b2 = (const float*)d_in[5];
    float* out = (float*)d_out;

    const int* src = ei;             // edge_index[0]
    const int* dst = ei + E_EDGES;   // edge_index[1]

    // workspace layout (floats)
    float* ws   = (float*)d_ws;
    float* dinv = ws;                                  // 50000   (deg -> dinv in place)
    float* h    = ws + 50176;                          // 50000*128
    float* agg  = ws + 50176 + 6400000;                // 50000*128
    float* h2   = ws + 50176 + 12800000;               // 50000*8

    const int TB = 256;

    GCN_init_deg <<<(N_NODES + TB - 1) / TB, TB, 0, stream>>>(dinv);
    GCN_count_deg<<<(E_EDGES + TB - 1) / TB, TB, 0, stream>>>(dst, dinv);
    GCN_rsqrt    <<<(N_NODES + TB - 1) / TB, TB, 0, stream>>>(dinv);

    GCN_gemm1<<<N_NODES / 16, TB, 0, stream>>>(x, W1, h);   // 3125 blocks, 8 waves each

    GCN_self1<<<(N_NODES * (F_HID / 4) + TB - 1) / TB, TB, 0, stream>>>(h, dinv, agg);
    GCN_edge1<<<(E_EDGES * 32 + TB - 1) / TB, TB, 0, stream>>>(src, dst, dinv, h, agg);

    GCN_gemm2<<<(N_NODES / 16 + 7) / 8, TB, 0, stream>>>(agg, b1, W2, h2);

    GCN_self2<<<(N_NODES * F_OUT + TB - 1) / TB, TB, 0, stream>>>(h2, dinv, b2, out);
    GCN_edge2<<<(E_EDGES + TB - 1) / TB, TB, 0, stream>>>(src, dst, dinv, h2, out);
}